// RankLP_66477503808086
// MI455X (gfx1250) — compile-verified
//
#include <hip/hip_runtime.h>
#include <hip/hip_bf16.h>

typedef float v2f __attribute__((ext_vector_type(2)));
typedef float v8f __attribute__((ext_vector_type(8)));

#define NN    20
#define NVAR  400      // n = N*N
#define MCON  440      // m = 2N + n
#define IPM_ITERS 25
#define TPB   256
#define LDAT  64       // row stride of padded KKT-core matrix
#define TROWS 48       // 40 real rows padded to 48

__device__ __forceinline__ float fair_of(int i) {
  // fair = tile(IDX_A/8 - (1-IDX_A)/12, N): depends only on column index
  return ((i % NN) < 8) ? 0.125f : (-1.0f / 12.0f);
}

__device__ __forceinline__ float block_sum(float v, float* red, int tid) {
  __syncthreads();
  red[tid] = v;
  __syncthreads();
  for (int off = TPB / 2; off > 0; off >>= 1) {
    if (tid < off) red[tid] += red[tid + off];
    __syncthreads();
  }
  float r = red[0];
  __syncthreads();
  return r;
}

__device__ __forceinline__ float block_min(float v, float* red, int tid) {
  __syncthreads();
  red[tid] = v;
  __syncthreads();
  for (int off = TPB / 2; off > 0; off >>= 1) {
    if (tid < off) red[tid] = fminf(red[tid], red[tid + off]);
    __syncthreads();
  }
  float r = red[0];
  __syncthreads();
  return r;
}

__global__ __launch_bounds__(TPB) void RankLP_ipm_kernel(const float* __restrict__ x,
                                                         float* __restrict__ out) {
  // Padded augmented Woodbury matrix: rows 0..39 = M, rows 40..47 = identity pad,
  // cols 48/49 = the two RHS vectors, cols 50..63 = zero pad (inert through LU).
  __shared__ float T[TROWS * LDAT];
  __shared__ float z[NVAR], p[NVAR], Dinv[NVAR], y1[NVAR], y2[NVAR], dz[NVAR], Hv2[NVAR];
  __shared__ float s[MCON], lam[MCON], rpri[MCON], d[MCON], tt[MCON], dsA[MCON], dlamA[MCON];
  __shared__ float red[TPB];
  __shared__ float rsA[NN], csA[NN], rs1[NN], cs1[NN], rs2[NN], cs2[NN], rsD[NN], csD[NN];
  __shared__ float scal[4];  // 0: nu, 1: pivinv, 2: x0, 3: x1

  const int tid = threadIdx.x;
  const int b = blockIdx.x;

  // ---- init: z = 0.5/N, s = h - G z (analytic), lam = 1, nu = 0, p = -x ----
  for (int i = tid; i < NVAR; i += TPB) { p[i] = -x[b * NVAR + i]; z[i] = 0.5f / NN; }
  for (int j = tid; j < MCON; j += TPB) { lam[j] = 1.0f; s[j] = (j < 2 * NN) ? 0.5f : 0.5f / NN; }
  if (tid == 0) scal[0] = 0.0f;
  __syncthreads();

  for (int it = 0; it < IPM_ITERS; ++it) {
    // ---- G z : row sums and column sums of z (grid N x N) ----
    if (tid < NN) {
      float a = 0; for (int c = 0; c < NN; ++c) a += z[tid * NN + c]; rsA[tid] = a;
    } else if (tid < 2 * NN) {
      int c = tid - NN; float a = 0; for (int r = 0; r < NN; ++r) a += z[r * NN + c]; csA[c] = a;
    }
    __syncthreads();
    // ---- r_pri = G z + s - h ----
    for (int j = tid; j < MCON; j += TPB) {
      float gz = (j < NN) ? rsA[j] : (j < 2 * NN) ? csA[j - NN] : -z[j - 2 * NN];
      float h = (j < 2 * NN) ? 1.0f : 0.0f;
      rpri[j] = gz + s[j] - h;
    }
    // ---- r_eq = fair . z ; mu = mean(s*lam) ----
    float pe = 0.0f, pm = 0.0f;
    for (int i = tid; i < NVAR; i += TPB) pe += fair_of(i) * z[i];
    for (int j = tid; j < MCON; j += TPB) pm += s[j] * lam[j];
    float req = block_sum(pe, red, tid);
    float mu = block_sum(pm, red, tid) * (1.0f / MCON);
    // ---- d = lam/s ; t = SIGMA*mu/s - lam + d*r_pri ----
    for (int j = tid; j < MCON; j += TPB) {
      float dd = lam[j] / s[j];
      d[j] = dd;
      tt[j] = 0.1f * mu / s[j] - lam[j] + dd * rpri[j];
    }
    __syncthreads();
    // ---- rhs_z = -(r_dual + G^T t) ; D^-1 ; Woodbury RHS vectors ----
    float nu = scal[0];
    for (int i = tid; i < NVAR; i += TPB) {
      int r = i / NN, c = i % NN;
      float gtl = lam[r] + lam[NN + c] - lam[2 * NN + i];      // (lam @ G)[i]
      float rd = 1e-4f * z[i] + p[i] + gtl + fair_of(i) * nu;  // r_dual
      float gtt = tt[r] + tt[NN + c] - tt[2 * NN + i];         // (t @ G)[i]
      float rz = -(rd + gtt);                                  // rhs_z
      float di = 1.0f / (1e-4f + d[2 * NN + i]);               // D^-1 (H diag part)
      Dinv[i] = di; y1[i] = di * rz; y2[i] = di * fair_of(i);
    }
    __syncthreads();
    // ---- u = E^T y (row/col sums) and row/col sums of D^-1 for M's diagonal ----
    if (tid < NN) {
      int r = tid; float a1 = 0, a2 = 0, aD = 0;
      for (int c = 0; c < NN; ++c) { a1 += y1[r * NN + c]; a2 += y2[r * NN + c]; aD += Dinv[r * NN + c]; }
      rs1[r] = a1; rs2[r] = a2; rsD[r] = aD;
    } else if (tid < 2 * NN) {
      int c = tid - NN; float a1 = 0, a2 = 0, aD = 0;
      for (int r = 0; r < NN; ++r) { a1 += y1[r * NN + c]; a2 += y2[r * NN + c]; aD += Dinv[r * NN + c]; }
      cs1[c] = a1; cs2[c] = a2; csD[c] = aD;
    }
    __syncthreads();
    // ---- build padded augmented M = W^-1 + E^T D^-1 E (SPD, 40x40 -> 48x48) ----
    for (int idx = tid; idx < TROWS * LDAT; idx += TPB) {
      int i = idx >> 6, jj = idx & 63;
      float v = 0.0f;
      if (jj < NN) {
        if (i < NN) v = (i == jj) ? (1.0f / d[i] + rsD[i]) : 0.0f;
        else if (i < 2 * NN) v = Dinv[jj * NN + (i - NN)];
      } else if (jj < 2 * NN) {
        if (i < NN) v = Dinv[i * NN + (jj - NN)];
        else if (i < 2 * NN) v = (i == jj) ? (1.0f / d[i] + csD[i - NN]) : 0.0f;
      } else if (jj == 48) {
        v = (i < NN) ? rs1[i] : (i < 2 * NN) ? cs1[i - NN] : 0.0f;
      } else if (jj == 49) {
        v = (i < NN) ? rs2[i] : (i < 2 * NN) ? cs2[i - NN] : 0.0f;
      } else if (jj == i) {
        v = 1.0f;  // identity padding rows 40..47
      }
      T[idx] = v;
    }
    // ---- blocked LU (no pivoting; M SPD) with fp32-WMMA Schur updates ----
    for (int kb = 0; kb < 3; ++kb) {
      const int bk = 16 * kb;
      for (int jc = bk; jc < bk + 16; ++jc) {
        __syncthreads();
        if (tid == 0) scal[1] = 1.0f / T[jc * LDAT + jc];
        __syncthreads();
        int i = jc + 1 + tid;  // one row per thread
        if (i < TROWS) {
          float lij = T[i * LDAT + jc] * scal[1];
          T[i * LDAT + jc] = lij;
          // rows inside block: full update (produces U12 + fwd-eliminated RHS);
          // rows below block: panel columns only (trailing done by WMMA Schur).
          int ce = (i <= bk + 15) ? LDAT : (bk + 16);
          for (int jj = jc + 1; jj < ce; ++jj)
            T[i * LDAT + jj] -= lij * T[jc * LDAT + jj];
        }
      }
      __syncthreads();
      if (bk + 16 < TROWS) {
        // Schur: A22 -= L21 * U12, K=16 via 4 chained V_WMMA_F32_16X16X4_F32
        const int t0 = bk + 16;
        const int nrt = (TROWS - t0) >> 4;
        const int nct = (LDAT - t0) >> 4;  // includes the RHS/pad column tile
        const int wv = tid >> 5;           // wave id (uniform in wave)
        const int lane = tid & 31;
        const int half = lane >> 4, lo = lane & 15;
        for (int tile = wv; tile < nrt * nct; tile += TPB / 32) {
          int r0 = t0 + 16 * (tile / nct);
          int c0 = t0 + 16 * (tile % nct);
          v8f acc;
#pragma unroll
          for (int jv = 0; jv < 8; ++jv)  // C/D: VGPR jv <-> M=jv / jv+8, lane=N
            acc[jv] = T[(r0 + jv + 8 * half) * LDAT + c0 + lo];
#pragma unroll
          for (int q = 0; q < 4; ++q) {
            int kk = bk + 4 * q + 2 * half;
            v2f af, bf;
            // A (16x4, lane=M): negate L21 so D = (-A)B + C computes C -= A*B
            af[0] = -T[(r0 + lo) * LDAT + kk];
            af[1] = -T[(r0 + lo) * LDAT + kk + 1];
            // B (4x16, lane=N)
            bf[0] = T[kk * LDAT + c0 + lo];
            bf[1] = T[(kk + 1) * LDAT + c0 + lo];
            acc = __builtin_amdgcn_wmma_f32_16x16x4_f32(false, af, false, bf,
                                                        (short)0, acc, false, false);
          }
#pragma unroll
          for (int jv = 0; jv < 8; ++jv)
            T[(r0 + jv + 8 * half) * LDAT + c0 + lo] = acc[jv];
        }
      }
      __syncthreads();
    }
    // ---- back substitution on the 40 real unknowns, both RHS at once ----
    for (int jc = 2 * NN - 1; jc >= 0; --jc) {
      __syncthreads();
      if (tid == 0) {
        float piv = 1.0f / T[jc * LDAT + jc];
        float x0 = T[jc * LDAT + 48] * piv;
        float x1 = T[jc * LDAT + 49] * piv;
        T[jc * LDAT + 48] = x0; T[jc * LDAT + 49] = x1;
        scal[2] = x0; scal[3] = x1;
      }
      __syncthreads();
      if (tid < jc) {
        T[tid * LDAT + 48] -= T[tid * LDAT + jc] * scal[2];
        T[tid * LDAT + 49] -= T[tid * LDAT + jc] * scal[3];
      }
    }
    __syncthreads();
    // ---- H^-1 v = D^-1 v - D^-1 E (M^-1 u) ; bordered elimination for nu ----
    float pAH1 = 0.0f, pAH2 = 0.0f;
    for (int i = tid; i < NVAR; i += TPB) {
      int r = i / NN, c = i % NN;
      float h1 = y1[i] - Dinv[i] * (T[r * LDAT + 48] + T[(NN + c) * LDAT + 48]);
      float h2 = y2[i] - Dinv[i] * (T[r * LDAT + 49] + T[(NN + c) * LDAT + 49]);
      y1[i] = h1; Hv2[i] = h2;
      pAH1 += fair_of(i) * h1; pAH2 += fair_of(i) * h2;
    }
    float AH1 = block_sum(pAH1, red, tid);
    float AH2 = block_sum(pAH2, red, tid);   // = A H^-1 A^T > 0
    float dnu = (AH1 + req) / AH2;
    for (int i = tid; i < NVAR; i += TPB) dz[i] = y1[i] - dnu * Hv2[i];
    __syncthreads();
    // ---- G dz row/col sums ----
    if (tid < NN) {
      float a = 0; for (int c = 0; c < NN; ++c) a += dz[tid * NN + c]; rs1[tid] = a;
    } else if (tid < 2 * NN) {
      int c = tid - NN; float a = 0; for (int r = 0; r < NN; ++r) a += dz[r * NN + c]; cs1[c] = a;
    }
    __syncthreads();
    // ---- ds, dlam, fraction-to-boundary step ----
    float pas = 1e30f, pal = 1e30f;
    for (int j = tid; j < MCON; j += TPB) {
      float gdz = (j < NN) ? rs1[j] : (j < 2 * NN) ? cs1[j - NN] : -dz[j - 2 * NN];
      float dsj = -rpri[j] - gdz;
      dsA[j] = dsj;
      float dlj = (0.1f * mu - s[j] * lam[j] - lam[j] * dsj) / s[j];
      dlamA[j] = dlj;
      if (dsj < 0.0f) pas = fminf(pas, -s[j] / dsj);
      if (dlj < 0.0f) pal = fminf(pal, -lam[j] / dlj);
    }
    float a_s = block_min(pas, red, tid);
    float a_l = block_min(pal, red, tid);
    float alpha = fminf(1.0f, 0.99f * fminf(a_s, a_l));
    // ---- updates ----
    for (int i = tid; i < NVAR; i += TPB) z[i] += alpha * dz[i];
    for (int j = tid; j < MCON; j += TPB) { s[j] += alpha * dsA[j]; lam[j] += alpha * dlamA[j]; }
    if (tid == 0) scal[0] += alpha * dnu;
    __syncthreads();
  }

  for (int i = tid; i < NVAR; i += TPB) out[b * NVAR + i] = z[i];
}

extern "C" void kernel_launch(void* const* d_in, const int* in_sizes, int n_in,
                              void* d_out, int out_size, void* d_ws, size_t ws_size,
                              hipStream_t stream) {
  (void)n_in; (void)out_size; (void)d_ws; (void)ws_size;
  const float* x = (const float*)d_in[0];
  float* out = (float*)d_out;
  int batch = in_sizes[0] / NVAR;  // 32
  RankLP_ipm_kernel<<<dim3(batch), dim3(TPB), 0, stream>>>(x, out);
}